// Police_17377437680144
// MI455X (gfx1250) — compile-verified
//
#include <hip/hip_runtime.h>
#include <hip/hip_bf16.h>
#include <cstdint>

// ---------------------------------------------------------------------------
// GATv2 x2 on MI455X (gfx1250, wave32).
// f32 WMMA (V_WMMA_F32_16X16X4_F32) for the two GEMMs, fused with the
// per-edge attention math. Node tensors (~60MB) are L2-resident (192MB L2);
// the edge_attr stream (205MB) is loaded non-temporally so it cannot evict
// them. B matrices staged transposed+padded in LDS so every B fragment is a
// single aligned ds_load_b64 (no repack movs).
// ---------------------------------------------------------------------------

typedef __attribute__((ext_vector_type(2))) float v2f;
typedef __attribute__((ext_vector_type(8))) float v8f;

#define WPB 4    // waves per block in WMMA kernels
#define TB  128  // their blockDim
#define WPAD 66  // LDS row pitch for 64-K weights (even -> 8B aligned v2f)
#define EPAD 34  // LDS row pitch for 32-K weights

__device__ __forceinline__ unsigned f2ord(float f) {
    unsigned u = __float_as_uint(f);
    return (u & 0x80000000u) ? ~u : (u | 0x80000000u);   // order-preserving map
}
__device__ __forceinline__ float ord2f(unsigned u) {
    return __uint_as_float((u & 0x80000000u) ? (u ^ 0x80000000u) : ~u);
}

// ---------------- small utility kernels ----------------
__global__ void k_fill_f32(float* p, int n, float v) {
    int t = blockIdx.x * blockDim.x + threadIdx.x;
    if (t < n) p[t] = v;
}
__global__ void k_fill_u32(unsigned* p, int n, unsigned v) {
    int t = blockIdx.x * blockDim.x + threadIdx.x;
    if (t < n) p[t] = v;
}
__global__ void k_build_aug(const int* __restrict__ src, const int* __restrict__ dst,
                            int* __restrict__ srcA, int* __restrict__ dstA,
                            int E, int Etot) {
    int t = blockIdx.x * blockDim.x + threadIdx.x;
    if (t >= Etot) return;
    if (t < E) { srcA[t] = src[t]; dstA[t] = dst[t]; }
    else       { srcA[t] = t - E;  dstA[t] = t - E; }   // self loops
}
// degree + sum of incoming edge_attr per dst (32 lanes per edge, coalesced)
__global__ void k_degloop(const int* __restrict__ dst, const float* __restrict__ eattr,
                          float* __restrict__ loopA, float* __restrict__ deg, int E) {
    long long t = (long long)blockIdx.x * blockDim.x + threadIdx.x;
    int e = (int)(t >> 5), k = (int)(t & 31);
    if (e >= E) return;
    int d = dst[e];
    float ea = __builtin_nontemporal_load(&eattr[(size_t)e * 32 + k]);  // NT: protect L2
    atomicAdd(&loopA[(size_t)d * 32 + k], ea);
    if (k == 0) atomicAdd(&deg[d], 1.0f);
}
__global__ void k_loopnorm(float* __restrict__ loopA, const float* __restrict__ deg, int N) {
    int t = blockIdx.x * blockDim.x + threadIdx.x;
    if (t >= N * 32) return;
    loopA[t] /= fmaxf(deg[t >> 5], 1.0f);
}

// ---------------- xl = x @ W + b  (f32 WMMA, one 16-row tile per wave) ----------------
__global__ void k_xw(const float* __restrict__ x, const float* __restrict__ W,
                     const float* __restrict__ b, float* __restrict__ y, int nNodes) {
    __shared__ float sW[64 * WPAD];   // transposed: sW[n*WPAD + k] = W[k*64 + n]
    __shared__ float sB[64];
    for (int i = threadIdx.x; i < 64 * 64; i += blockDim.x) {
        int k = i >> 6, n = i & 63;
        sW[n * WPAD + k] = W[i];      // coalesced read; conflict-free strided LDS write
    }
    if (threadIdx.x < 64) sB[threadIdx.x] = b[threadIdx.x];
    __syncthreads();

    int wave = threadIdx.x >> 5, lane = threadIdx.x & 31;
    int tile = blockIdx.x * WPB + wave;
    int nTiles = (nNodes + 15) >> 4;
    if (tile >= nTiles) return;                    // wave-uniform: EXEC stays all-1s

    int row = lane & 15;
    int kh  = (lane >> 4) << 1;                    // K offset 0 or 2 per A-frag layout
    int nr  = tile * 16 + row; if (nr >= nNodes) nr = nNodes - 1;
    const float* xrow = x + (size_t)nr * 64;

    v8f acc[4] = {};
#pragma unroll
    for (int k0 = 0; k0 < 64; k0 += 4) {
        v2f a = *(const v2f*)(xrow + k0 + kh);
#pragma unroll
        for (int nt = 0; nt < 4; ++nt) {
            v2f bb = *(const v2f*)&sW[(nt * 16 + row) * WPAD + k0 + kh]; // one ds_load_b64
            acc[nt] = __builtin_amdgcn_wmma_f32_16x16x4_f32(
                false, a, false, bb, (short)0, acc[nt], false, false);
        }
    }
    int em = (lane >> 4) << 3, fl = lane & 15;
#pragma unroll
    for (int r = 0; r < 8; ++r) {
        int node = tile * 16 + em + r;
        if (node < nNodes) {
#pragma unroll
            for (int nt = 0; nt < 4; ++nt) {
                int f = nt * 16 + fl;
                y[(size_t)node * 64 + f] = acc[nt][r] + sB[f];
            }
        }
    }
}

// -------- fused: m = leaky(xl[src]+xl[dst]+eattr@We); logit = m.att; segmax --------
__global__ void k_msg(const float* __restrict__ eattr, const float* __restrict__ loopA,
                      const int* __restrict__ srcA, const int* __restrict__ dstA,
                      const float* __restrict__ xl, const float* __restrict__ We,
                      const float* __restrict__ att,
                      float* __restrict__ logit, unsigned* __restrict__ mx,
                      int E, int Etot) {
    __shared__ float sWe[64 * EPAD]; // transposed: sWe[n*EPAD + k] = We[k*64 + n]
    __shared__ float sAtt[64];
    for (int i = threadIdx.x; i < 32 * 64; i += blockDim.x) {
        int k = i >> 6, n = i & 63;
        sWe[n * EPAD + k] = We[i];
    }
    if (threadIdx.x < 64) sAtt[threadIdx.x] = att[threadIdx.x];
    __syncthreads();

    int wave = threadIdx.x >> 5, lane = threadIdx.x & 31;
    int tile = blockIdx.x * WPB + wave;
    int nTiles = (Etot + 15) >> 4;
    if (tile >= nTiles) return;                    // wave-uniform exit

    int row = lane & 15;
    int kh  = (lane >> 4) << 1;
    int eA  = tile * 16 + row; if (eA >= Etot) eA = Etot - 1;
    bool isReal = (eA < E);
    const float* erow = isReal ? (eattr + (size_t)eA * 32)
                               : (loopA + (size_t)(eA - E) * 32);

    v8f acc[4] = {};
#pragma unroll
    for (int k0 = 0; k0 < 32; k0 += 4) {
        // NT load: edge_attr is a 205MB one-shot stream; keep it out of L2 so the
        // node tensors (xl/out/mx/denom) stay resident.
        v2f a;
        if (isReal) a = __builtin_nontemporal_load((const v2f*)(erow + k0 + kh));
        else        a = *(const v2f*)(erow + k0 + kh);
#pragma unroll
        for (int nt = 0; nt < 4; ++nt) {
            v2f bb = *(const v2f*)&sWe[(nt * 16 + row) * EPAD + k0 + kh];
            acc[nt] = __builtin_amdgcn_wmma_f32_16x16x4_f32(
                false, a, false, bb, (short)0, acc[nt], false, false);
        }
    }

    // D layout: VGPR r, lanes 0-15 -> edge r, lanes 16-31 -> edge r+8; column = lane&15
    int em = (lane >> 4) << 3, fl = lane & 15;
#pragma unroll
    for (int r = 0; r < 8; ++r) {
        int ee = tile * 16 + em + r;
        float part = 0.f;
        if (ee < Etot) {
            int s = srcA[ee], d = dstA[ee];
#pragma unroll
            for (int nt = 0; nt < 4; ++nt) {
                int f = nt * 16 + fl;
                float xv = xl[(size_t)s * 64 + f] + xl[(size_t)d * 64 + f]; // 64B coalesced per half-wave
                float m  = acc[nt][r] + xv;
                m = (m > 0.f) ? m : 0.2f * m;     // leaky relu
                part += m * sAtt[f];
            }
        }
        // reduce the 16 column-lanes of this half-wave
        part += __shfl_xor(part, 1, 32);
        part += __shfl_xor(part, 2, 32);
        part += __shfl_xor(part, 4, 32);
        part += __shfl_xor(part, 8, 32);
        if (fl == 0 && ee < Etot) {
            logit[ee] = part;
            atomicMax(&mx[dstA[ee]], f2ord(part));
        }
    }
}

__global__ void k_expdenom(const float* __restrict__ logit, const int* __restrict__ dstA,
                           const unsigned* __restrict__ mx, float* __restrict__ ew,
                           float* __restrict__ denom, int Etot) {
    int e = blockIdx.x * blockDim.x + threadIdx.x;
    if (e >= Etot) return;
    int d = dstA[e];
    float w = expf(logit[e] - ord2f(mx[d]));
    ew[e] = w;
    atomicAdd(&denom[d], w);
}

__global__ void k_initout(float* __restrict__ out, const float* __restrict__ bias, int N) {
    int t = blockIdx.x * blockDim.x + threadIdx.x;
    if (t >= N * 64) return;
    out[t] = bias[t & 63];
}

__global__ void k_scatter(const int* __restrict__ srcA, const int* __restrict__ dstA,
                          const float* __restrict__ ew, const float* __restrict__ denom,
                          const float* __restrict__ xl, float* __restrict__ out, int Etot) {
    long long t = (long long)blockIdx.x * blockDim.x + threadIdx.x;
    int e = (int)(t >> 6), f = (int)(t & 63);
    if (e >= Etot) return;
    int s = srcA[e], d = dstA[e];
    float alpha = ew[e] / denom[d];
    atomicAdd(&out[(size_t)d * 64 + f], xl[(size_t)s * 64 + f] * alpha);
}

// ---------------------------------------------------------------------------
static inline int cdivll(long long a, long long b) { return (int)((a + b - 1) / b); }

extern "C" void kernel_launch(void* const* d_in, const int* in_sizes, int n_in,
                              void* d_out, int out_size, void* d_ws, size_t ws_size,
                              hipStream_t stream) {
    const float* x     = (const float*)d_in[0];
    const int*   ei    = (const int*)  d_in[1];
    const float* eattr = (const float*)d_in[2];
    const float* W1    = (const float*)d_in[3];
    const float* b1    = (const float*)d_in[4];
    const float* We1   = (const float*)d_in[5];
    const float* att1  = (const float*)d_in[6];
    const float* bias1 = (const float*)d_in[7];
    const float* W2    = (const float*)d_in[8];
    const float* b2    = (const float*)d_in[9];
    const float* We2   = (const float*)d_in[10];
    const float* att2  = (const float*)d_in[11];
    const float* bias2 = (const float*)d_in[12];

    const int N    = in_sizes[0] / 64;
    const int E    = in_sizes[1] / 2;
    const int Etot = E + N;
    const int* src = ei;
    const int* dst = ei + E;

    // workspace carve-up (256B aligned)
    char* w = (char*)d_ws; size_t off = 0;
    auto take = [&](size_t bytes) -> void* {
        void* p = w + off; off += (bytes + 255) & ~(size_t)255; return p;
    };
    int*      srcA  = (int*)     take(sizeof(int) * (size_t)Etot);
    int*      dstA  = (int*)     take(sizeof(int) * (size_t)Etot);
    float*    loopA = (float*)   take(sizeof(float) * (size_t)N * 32);
    float*    deg   = (float*)   take(sizeof(float) * (size_t)N);
    float*    xl    = (float*)   take(sizeof(float) * (size_t)N * 64);
    float*    h     = (float*)   take(sizeof(float) * (size_t)N * 64);
    float*    logit = (float*)   take(sizeof(float) * (size_t)Etot);
    float*    ew    = (float*)   take(sizeof(float) * (size_t)Etot);
    unsigned* mx    = (unsigned*)take(sizeof(unsigned) * (size_t)N);
    float*    denom = (float*)   take(sizeof(float) * (size_t)N);
    (void)ws_size; (void)n_in; (void)out_size;

    // --- shared setup: self-loop attrs (mean of incoming), augmented indices ---
    k_fill_f32<<<cdivll((long long)N * 32, 256), 256, 0, stream>>>(loopA, N * 32, 0.f);
    k_fill_f32<<<cdivll(N, 256), 256, 0, stream>>>(deg, N, 0.f);
    k_build_aug<<<cdivll(Etot, 256), 256, 0, stream>>>(src, dst, srcA, dstA, E, Etot);
    k_degloop<<<cdivll((long long)E * 32, 256), 256, 0, stream>>>(dst, eattr, loopA, deg, E);
    k_loopnorm<<<cdivll((long long)N * 32, 256), 256, 0, stream>>>(loopA, deg, N);

    const int nTilesN = (N + 15) / 16;
    const int nTilesE = (Etot + 15) / 16;
    const unsigned ORD_NEG_INF = 0x007FFFFFu;   // f2ord(-inf)

    auto layer = [&](const float* xin, const float* W, const float* bb,
                     const float* We, const float* att, const float* bias, float* out) {
        k_xw<<<cdivll(nTilesN, WPB), TB, 0, stream>>>(xin, W, bb, xl, N);
        k_fill_u32<<<cdivll(N, 256), 256, 0, stream>>>(mx, N, ORD_NEG_INF);
        k_fill_f32<<<cdivll(N, 256), 256, 0, stream>>>(denom, N, 0.f);
        k_msg<<<cdivll(nTilesE, WPB), TB, 0, stream>>>(eattr, loopA, srcA, dstA, xl,
                                                       We, att, logit, mx, E, Etot);
        k_expdenom<<<cdivll(Etot, 256), 256, 0, stream>>>(logit, dstA, mx, ew, denom, Etot);
        k_initout<<<cdivll((long long)N * 64, 256), 256, 0, stream>>>(out, bias, N);
        k_scatter<<<cdivll((long long)Etot * 64, 256), 256, 0, stream>>>(srcA, dstA, ew,
                                                                         denom, xl, out, Etot);
    };

    layer(x, W1, b1, We1, att1, bias1, h);
    layer(h, W2, b2, We2, att2, bias2, (float*)d_out);
}